// AdSBHNet_523986010335
// MI455X (gfx1250) — compile-verified
//
#include <hip/hip_runtime.h>
#include <hip/hip_bf16.h>

typedef __attribute__((ext_vector_type(2))) float v2f;
typedef __attribute__((ext_vector_type(8))) float v8f;

namespace {

__device__ __forceinline__ float frcp(float x)  { return __builtin_amdgcn_rcpf(x);  }
__device__ __forceinline__ float frsq(float x)  { return __builtin_amdgcn_rsqf(x);  }
__device__ __forceinline__ float fsqrt_(float x){ return __builtin_amdgcn_sqrtf(x); }

constexpr float PI_F = 3.14159265358979323846f;

// f(z) = sum_i 4 c_i (z^4 - z^i)/(i-4), c = [1, a0, a1]  (i in {0,1,2}, no log term)
__device__ __forceinline__ float eval_f_dev(float a0, float a1, float z, float z2, float z4) {
  return (1.0f - z4) - (4.0f / 3.0f) * a0 * (z4 - z) - 2.0f * a1 * (z4 - z2);
}

} // namespace

// One wave handles a tile of 16 zs values. Each lane evaluates 2 quadrature
// steps per chunk (matching the documented 16x4 f32 A-matrix layout); the
// trapz (incl. y->0 extrapolation and endpoint extension, folded into per-step
// weights) is performed on the matrix pipe via V_WMMA_F32_16X16X4_F32 with an
// all-ones B matrix:  D[m][*] += sum_k w_k * integrand[m,k].
__global__ void __launch_bounds__(32) adsbh_wmma_kernel(
    const float* __restrict__ pa, const float* __restrict__ pb,
    const float* __restrict__ plogc, const float* __restrict__ pshift,
    const float* __restrict__ zs, float* __restrict__ out, int nB)
{
  const int lane = threadIdx.x & 31;
  const int tile = blockIdx.x;
  const int m    = lane & 15;            // zs row this lane feeds into A
  const int gm   = tile * 16 + m;

  const float a0 = pa[0], a1 = pa[1];
  const float b0 = pb[0], b1 = pb[1];
  const float zsv = zs[gm];

  // fs = eval_f(a, zs)
  const float zs2 = zsv * zsv, zs4 = zs2 * zs2;
  const float fs     = eval_f_dev(a0, a1, zsv, zs2, zs4);
  const float inv_fs = frcp(fs);

  // --- quadrature weights (trapz + extrapolation to 0 + endpoint extension) ---
  // grid 1: y = linspace(0.001, 0.999, 1000)
  const float dy   = 0.998f / 999.0f;
  const float r1   = 0.001f / dy;
  const float W0   = 0.5f * 0.001f * (2.0f + r1) + 0.5f * dy;  // f0 coeff
  const float W1   = dy - 0.5f * 0.001f * r1;                  // f1 coeff
  const float W999 = 0.5f * dy + 0.5f * 0.001f;                // f999 coeff
  // grid 2: y2 = linspace(0.001, 1.0, 1000)  (dy2 = 0.999/999 = 0.001 exactly)
  const float dy2   = 0.001f;
  const float W0d   = 0.002f;   // 0.5*1e-3*(2+1) + 0.5*1e-3
  const float W1d   = 0.0005f;  // 1e-3 - 0.5*1e-3
  const float W999d = 0.0005f;  // 0.5*1e-3

  // A-matrix 16x4 f32 layout: lanes 0-15 -> K = {0,1}; lanes 16-31 -> K = {2,3}
  const int koff = (lane >> 4) << 1;

  v8f accL = {};  // L integral accumulator   (16x16 f32 D/C)
  v8f accV = {};  // V_connected accumulator
  v8f accD = {};  // V_disconnected accumulator
  v2f bones = {1.0f, 1.0f};  // B = all ones (layout independent)

  for (int chunk = 0; chunk < 250; ++chunk) {
    v2f aL, aV, aD;
#pragma unroll
    for (int j = 0; j < 2; ++j) {
      const int   s  = chunk * 4 + koff + j;   // quadrature step index 0..999
      const float fy = (float)s;

      // ---------- grid 1: y in [0.001, 0.999] (L and V_connected) ----------
      const float y  = fmaf(fy, dy, 0.001f);
      const float u  = 1.0f - y, v = 1.0f + y;
      const float w  = u * v;                  // (1-y)(1+y)
      const float z  = zsv * w;
      const float z2 = z * z, z4 = z2 * z2;
      const float f   = eval_f_dev(a0, a1, z, z2, z4);
      const float bz  = fmaf(b1, z2, fmaf(b0, z, 1.0f));
      const float g   = bz * frcp(1.0f - z4);  // eval_g
      const float fof = f * inv_fs;            // f / fs
      const float w2  = w * w, w4 = w2 * w2;   // (1-y)^4 (1+y)^4

      // L: sqrt(g) / sqrt(f_over_fs / w^4 - 1) * y
      const float iL = fsqrt_(g) * frsq(fof * frcp(w4) - 1.0f) * y;
      // V_c: sqrt(f*g)/w^2 * (1/sqrt(1 - w^4 / f_over_fs) - 1) * y
      const float iV = fsqrt_(f * g) * frcp(w2) *
                       (frsq(1.0f - w4 * frcp(fof)) - 1.0f) * y;

      float wk = dy;
      if (s == 0) wk = W0; else if (s == 1) wk = W1; else if (s == 999) wk = W999;
      aL[j] = wk * iL;
      aV[j] = wk * iV;

      // ---------- grid 2: y2 in [0.001, 1.0] (V_disconnected) ----------
      const float y2  = fmaf(fy, dy2, 0.001f);
      const float zd  = 1.0f - (1.0f - zsv) * y2;
      const float zd2 = zd * zd, zd4 = zd2 * zd2;
      const float fd  = eval_f_dev(a0, a1, zd, zd2, zd4);
      const float bzd = fmaf(b1, zd2, fmaf(b0, zd, 1.0f));
      const float gd  = bzd * frcp(1.0f - zd4);
      const float iD  = fsqrt_(fd * gd) * frcp(zd2);

      float wkd = dy2;
      if (s == 0) wkd = W0d; else if (s == 1) wkd = W1d; else if (s == 999) wkd = W999d;
      aD[j] = wkd * iD;
    }

    // D = A x B + C ; B all-ones -> every column of D accumulates sum_k w_k f_k
    accL = __builtin_amdgcn_wmma_f32_16x16x4_f32(false, aL, false, bones,
                                                 (short)0, accL, false, false);
    accV = __builtin_amdgcn_wmma_f32_16x16x4_f32(false, aV, false, bones,
                                                 (short)0, accV, false, false);
    accD = __builtin_amdgcn_wmma_f32_16x16x4_f32(false, aD, false, bones,
                                                 (short)0, accD, false, false);
  }

  // 16x16 f32 D layout: VGPR r -> M=r on lanes 0-15, M=8+r on lanes 16-31.
  // All columns identical, so lane 0 holds M=0..7, lane 16 holds M=8..15.
  if (lane == 0 || lane == 16) {
    const int   mbase = tile * 16 + ((lane == 16) ? 8 : 0);
    const float ec = expf(plogc[0]);
    const float sh = pshift[0];
#pragma unroll
    for (int r = 0; r < 8; ++r) {
      const int   mg = mbase + r;
      const float z  = zs[mg];
      const float Lv = 4.0f * z * accL[r] * (1.0f / PI_F);
      const float Vc = 4.0f * PI_F * accV[r] * frcp(z);
      const float Vd = 2.0f * PI_F * (1.0f - z) * accD[r];
      out[mg]      = Lv;                      // row 0: L
      out[nB + mg] = fmaf(ec, Vc - Vd, sh);   // row 1: V
    }
  }
}

extern "C" void kernel_launch(void* const* d_in, const int* in_sizes, int n_in,
                              void* d_out, int out_size, void* d_ws, size_t ws_size,
                              hipStream_t stream) {
  // setup_inputs() order: a[2], b[2], logcoef[1], shift[1], zs[B]  (all float32)
  const float* a     = (const float*)d_in[0];
  const float* b     = (const float*)d_in[1];
  const float* logc  = (const float*)d_in[2];
  const float* shift = (const float*)d_in[3];
  const float* zsp   = (const float*)d_in[4];
  float* out = (float*)d_out;

  const int nB    = in_sizes[4];   // 8192
  const int tiles = nB / 16;       // 16 zs per wave, one wave per block

  adsbh_wmma_kernel<<<tiles, 32, 0, stream>>>(a, b, logc, shift, zsp, out, nB);
}